// RTRL_73160472920108
// MI455X (gfx1250) — compile-verified
//
#include <hip/hip_runtime.h>
#include <cmath>

// Problem sizes (match reference): H=192, I=96, O=48, T=16
constexpr int Hd = 192;
constexpr int Id = 96;
constexpr int Od = 48;
constexpr int Td = 16;

// d_out flat layout (reference return order):
// acc_loss(1), y_hats(T*O), gWhh(H*H), gWxh(H*I), gb(H), gWout(O*H), gbout(O)
constexpr int OFF_LOSS  = 0;
constexpr int OFF_YHAT  = 1;
constexpr int OFF_GWHH  = OFF_YHAT  + Td * Od;   // 769
constexpr int OFF_GWXH  = OFF_GWHH  + Hd * Hd;   // 37633
constexpr int OFF_GB    = OFF_GWXH  + Hd * Id;   // 56065
constexpr int OFF_GWOUT = OFF_GB    + Hd;        // 56257
constexpr int OFF_GBOUT = OFF_GWOUT + Od * Hd;   // 65473

typedef __attribute__((ext_vector_type(2))) float v2f;
typedef __attribute__((ext_vector_type(8))) float v8f;

// ---------------------------------------------------------------------------
// Phase 1: sequential forward pass. One workgroup, 192 threads (6 wave32).
// Computes h_t, y_hat_t, loss, gy_t = dL/dy_hat_t, gh_t = W_out^T gy_t.
// Also materializes Hprev[t] = h_{t-1} (Hprev[0] = 0) so the gradient GEMM
// kernel has a branch-free B operand for gWhh.
// ---------------------------------------------------------------------------
__global__ __launch_bounds__(Hd) void rtrl_fwd(
    const float* __restrict__ x, const float* __restrict__ tgt,
    const float* __restrict__ mask,
    const float* __restrict__ Whh, const float* __restrict__ Wxh,
    const float* __restrict__ bh, const float* __restrict__ Wout,
    const float* __restrict__ bout,
    float* __restrict__ out,
    float* __restrict__ Hmat, float* __restrict__ Hprev,
    float* __restrict__ Gh, float* __restrict__ Gy)
{
    __shared__ float hs[Hd];
    __shared__ float gys[Od];
    __shared__ float lossAcc;
    const int i = threadIdx.x;
    hs[i] = 0.0f;
    Hprev[i] = 0.0f;                   // h_{-1} = 0
    if (i == 0) lossAcc = 0.0f;
    __syncthreads();

    for (int t = 0; t < Td; ++t) {
        // pre_i = (W_hh h_{t-1})_i + (W_xh x_t)_i + b_i
        float acc = bh[i];
        const float* wr = Whh + i * Hd;
        #pragma unroll 4
        for (int j = 0; j < Hd; ++j) acc += wr[j] * hs[j];
        const float* wx = Wxh + i * Id;
        const float* xr = x + t * Id;
        #pragma unroll 4
        for (int j = 0; j < Id; ++j) acc += wx[j] * xr[j];
        float ht = tanhf(acc);

        __syncthreads();               // everyone done reading old hs
        hs[i] = ht;
        Hmat[t * Hd + i] = ht;
        if (t + 1 < Td) Hprev[(t + 1) * Hd + i] = ht;
        __syncthreads();               // new hs visible

        // readout + loss + gy
        if (i < Od) {
            float yh = bout[i];
            const float* wo = Wout + i * Hd;
            #pragma unroll 4
            for (int j = 0; j < Hd; ++j) yh += wo[j] * hs[j];
            float e = yh - tgt[t * Od + i];
            float m = mask[t];
            float g = 2.0f * m * e;
            gys[i] = g;
            Gy[t * Od + i] = g;
            out[OFF_YHAT + t * Od + i] = yh;
            atomicAdd(&lossAcc, m * e * e);
        }
        __syncthreads();               // gys visible

        // gh_t = W_out^T gy_t
        float gh = 0.0f;
        #pragma unroll 4
        for (int o = 0; o < Od; ++o) gh += Wout[o * Hd + i] * gys[o];
        Gh[t * Hd + i] = gh;
    }
    __syncthreads();
    if (i == 0) out[OFF_LOSS] = lossAcc;
}

// ---------------------------------------------------------------------------
// Phase 2: sequential backward (adjoint) recurrence.
// delta_t = (1 - h_t^2) .* (gh_t + W_hh^T delta_{t+1});  gb = sum_t delta_t;
// gbout = sum_t gy_t.  One workgroup, 192 threads.
// ---------------------------------------------------------------------------
__global__ __launch_bounds__(Hd) void rtrl_bwd(
    const float* __restrict__ Whh,
    const float* __restrict__ Hmat, const float* __restrict__ Gh,
    const float* __restrict__ Gy,
    float* __restrict__ out, float* __restrict__ Delta)
{
    __shared__ float ds[Hd];
    const int i = threadIdx.x;
    ds[i] = 0.0f;
    float gb = 0.0f;
    __syncthreads();

    for (int t = Td - 1; t >= 0; --t) {
        float acc = Gh[t * Hd + i];
        #pragma unroll 4
        for (int j = 0; j < Hd; ++j) acc += Whh[j * Hd + i] * ds[j];
        float ht = Hmat[t * Hd + i];
        float d = (1.0f - ht * ht) * acc;
        __syncthreads();               // everyone done reading old ds
        ds[i] = d;
        __syncthreads();               // new ds visible
        Delta[t * Hd + i] = d;
        gb += d;
    }
    out[OFF_GB + i] = gb;
    if (i < Od) {
        float s = 0.0f;
        #pragma unroll
        for (int t = 0; t < Td; ++t) s += Gy[t * Od + i];
        out[OFF_GBOUT + i] = s;
    }
}

// ---------------------------------------------------------------------------
// Phase 3: gradient accumulation GEMMs via V_WMMA_F32_16X16X4_F32.
//   gWhh  = Delta^T (192x16) @ Hprev (16x192)  -> 12x12 tiles
//   gWxh  = Delta^T (192x16) @ X     (16x96)   -> 12x6  tiles
//   gWout = Gy^T    (48x16)  @ Hmat  (16x192)  -> 3x12  tiles
// One wave per 16x16 output tile; K = T = 16 => four K=4 WMMA ops per tile.
// All mode-dependent bases/strides are selected uniformly BEFORE the K-loop,
// so the inner loop is 4 branch-free {4 loads + 1 wmma} groups.
// ISA A-layout (32-bit A 16x4): lanes 0-15 hold K={k0,k0+1} in v0/v1, lanes
// 16-31 hold K={k0+2,k0+3}; B mirrors with N across lanes. C/D: VGPR v,
// lanes 0-15 row M=v, lanes 16-31 row M=8+v.
// ---------------------------------------------------------------------------
__global__ __launch_bounds__(32) void rtrl_grads(
    const float* __restrict__ Delta, const float* __restrict__ Hmat,
    const float* __restrict__ Hprev, const float* __restrict__ Gy,
    const float* __restrict__ x, float* __restrict__ out)
{
    const int tile = blockIdx.x;
    const int lane = threadIdx.x;
    const int half = lane >> 4;      // 0: K pair {k0,k0+1}, 1: {k0+2,k0+3}
    const int lm   = lane & 15;      // M index (A) / N index (B)

    // Uniform (per-block) operand selection, hoisted out of the K-loop.
    const float* A; int sA;          // A[k*sA + m]  (time-major activations)
    const float* B; int sB;          // B[k*sB + n]
    float* C; int ldc; int m0, n0;
    if (tile < 144) {                // gWhh = Delta^T @ Hprev
        m0 = (tile / 12) * 16; n0 = (tile % 12) * 16;
        A = Delta; sA = Hd; B = Hprev; sB = Hd;
        C = out + OFF_GWHH; ldc = Hd;
    } else if (tile < 216) {         // gWxh = Delta^T @ X
        const int u = tile - 144;
        m0 = (u / 6) * 16; n0 = (u % 6) * 16;
        A = Delta; sA = Hd; B = x; sB = Id;
        C = out + OFF_GWXH; ldc = Id;
    } else {                         // gWout = Gy^T @ Hmat
        const int u = tile - 216;
        m0 = (u / 12) * 16; n0 = (u % 12) * 16;
        A = Gy; sA = Od; B = Hmat; sB = Hd;
        C = out + OFF_GWOUT; ldc = Hd;
    }
    const float* Ap = A + m0 + lm;   // per-lane column of A^T tile
    const float* Bp = B + n0 + lm;   // per-lane column of B tile

    v8f acc = {0.f, 0.f, 0.f, 0.f, 0.f, 0.f, 0.f, 0.f};

    #pragma unroll
    for (int c = 0; c < 4; ++c) {
        const int k0 = 4 * c + 2 * half;   // this lane-half's starting K
        v2f a = { Ap[k0 * sA], Ap[(k0 + 1) * sA] };
        v2f b = { Bp[k0 * sB], Bp[(k0 + 1) * sB] };
        // 8 args: (neg_a, A, neg_b, B, c_mod, C, reuse_a, reuse_b)
        acc = __builtin_amdgcn_wmma_f32_16x16x4_f32(
            false, a, false, b, (short)0, acc, false, false);
    }

    #pragma unroll
    for (int v = 0; v < 8; ++v) {
        const int row = m0 + v + 8 * half;
        C[row * ldc + n0 + lm] = acc[v];
    }
}

// ---------------------------------------------------------------------------
extern "C" void kernel_launch(void* const* d_in, const int* in_sizes, int n_in,
                              void* d_out, int out_size, void* d_ws, size_t ws_size,
                              hipStream_t stream) {
    (void)in_sizes; (void)n_in; (void)out_size; (void)ws_size;
    const float* x    = (const float*)d_in[0];  // [T,I]
    const float* tgt  = (const float*)d_in[1];  // [T,O]
    const float* mask = (const float*)d_in[2];  // [T]
    const float* Whh  = (const float*)d_in[3];  // [H,H]
    const float* Wxh  = (const float*)d_in[4];  // [H,I]
    const float* bh   = (const float*)d_in[5];  // [H]
    const float* Wout = (const float*)d_in[6];  // [O,H]
    const float* bout = (const float*)d_in[7];  // [O]
    float* out = (float*)d_out;

    float* ws    = (float*)d_ws;
    float* Hmat  = ws;                 // [T,H]
    float* Hprev = Hmat  + Td * Hd;    // [T,H]  (h_{t-1}, row 0 = zeros)
    float* Gh    = Hprev + Td * Hd;    // [T,H]
    float* Gy    = Gh    + Td * Hd;    // [T,O]
    float* Delta = Gy    + Td * Od;    // [T,H]

    hipLaunchKernelGGL(rtrl_fwd, dim3(1), dim3(Hd), 0, stream,
                       x, tgt, mask, Whh, Wxh, bh, Wout, bout, out,
                       Hmat, Hprev, Gh, Gy);
    hipLaunchKernelGGL(rtrl_bwd, dim3(1), dim3(Hd), 0, stream,
                       Whh, Hmat, Gh, Gy, out, Delta);
    hipLaunchKernelGGL(rtrl_grads, dim3(252), dim3(32), 0, stream,
                       Delta, Hmat, Hprev, Gy, x, out);
}